// SoftEIGNN_31044023616077
// MI455X (gfx1250) — compile-verified
//
#include <hip/hip_runtime.h>
#include <stdint.h>

#define DD 128
#define KAPPA 0.95f
#define FEPS 1e-5f

typedef __attribute__((ext_vector_type(2))) float v2f;
typedef __attribute__((ext_vector_type(8))) float v8f;

// Native, non-returning f32 atomic add (relaxed, device scope) ->
// global_atomic_add_f32 on gfx1250 (L2 fp atomics are no-flush per ISA ch12).
__device__ __forceinline__ void atomAddF32(float* p, float v) {
    __hip_atomic_fetch_add(p, v, __ATOMIC_RELAXED, __HIP_MEMORY_SCOPE_AGENT);
}

// ---------------------------------------------------------------------------
// 1) acc1 (= d_out) <- bias broadcast ; acc2 <- 0 ; ss <- 0
// ---------------------------------------------------------------------------
__global__ void init_acc(const float* __restrict__ bias,
                         float* __restrict__ acc1,
                         float* __restrict__ acc2,
                         float* __restrict__ ss, int total) {
    int i = blockIdx.x * 256 + threadIdx.x;
    if (i == 0) *ss = 0.0f;
    if (i < total) {
        acc1[i] = bias[i & (DD - 1)];
        acc2[i] = 0.0f;
    }
}

// ---------------------------------------------------------------------------
// 2) Wraw[i][j] = sum_k F[k][i]*F[k][j]; atomic-accumulate sum of squares
//    (128x128 output, 2 MFLOP total -> plain VALU is the right tool)
// ---------------------------------------------------------------------------
__global__ void wmat(const float* __restrict__ F,
                     float* __restrict__ Wr,
                     float* __restrict__ ss) {
    int idx = blockIdx.x * 256 + threadIdx.x;    // 0..16383
    int i = idx >> 7, j = idx & 127;
    float acc = 0.0f;
    for (int k = 0; k < DD; ++k)
        acc += F[k * DD + i] * F[k * DD + j];
    Wr[idx] = acc;
    atomAddF32(ss, acc * acc);
}

// ---------------------------------------------------------------------------
// 3) Wk = KAPPA * Wraw / (sqrt(ss) + eps)   (in place)
// ---------------------------------------------------------------------------
__global__ void wscale(float* __restrict__ Wk, const float* __restrict__ ss) {
    int idx = blockIdx.x * 256 + threadIdx.x;
    float s = KAPPA / (sqrtf(*ss) + FEPS);
    Wk[idx] *= s;
}

// ---------------------------------------------------------------------------
// 4) xW[n][j] = sum_k features[n][k] * gw[j][k]   (WMMA f32 16x16x4)
//    One 16x16 output tile per wave; 4 waves per block share the same 16
//    A-rows (consecutive n-tiles) for L0 reuse. A fragment: contiguous
//    float2 per lane at A[m0+lid][k + 2*half]; B = gw^T so its fragment is
//    also a contiguous float2 from gw's row (col*128 + k + 2*half).
// ---------------------------------------------------------------------------
__global__ void gemm_xw(const float* __restrict__ A,
                        const float* __restrict__ Gw,
                        float* __restrict__ Cout, int ntiles) {
    int wave = blockIdx.x * (blockDim.x >> 5) + (threadIdx.x >> 5);
    if (wave >= ntiles) return;                  // wave-uniform -> EXEC stays all-1s
    int lane = threadIdx.x & 31;
    int mt = wave >> 3;                          // 8 column tiles (D=128)
    int nt = wave & 7;
    int half = lane >> 4, lid = lane & 15;

    const float* Ap = A  + (size_t)(mt * 16 + lid) * DD + 2 * half;
    const float* Bp = Gw + (size_t)(nt * 16 + lid) * DD + 2 * half;

    v8f c = {};
#pragma unroll
    for (int k = 0; k < DD; k += 4) {
        v2f a = *(const v2f*)(Ap + k);
        v2f b = *(const v2f*)(Bp + k);
        c = __builtin_amdgcn_wmma_f32_16x16x4_f32(false, a, false, b,
                                                  (short)0, c, false, false);
    }
    float* Op = Cout + (size_t)(mt * 16 + 8 * half) * DD + nt * 16 + lid;
#pragma unroll
    for (int v = 0; v < 8; ++v)
        Op[(size_t)v * DD] = c[v];
}

// ---------------------------------------------------------------------------
// 5) Fused double SpMM: one wave per edge; float4 gather from xW and emb
//    (both L2-resident), 8 native global_atomic_add_f32 per lane.
// ---------------------------------------------------------------------------
__global__ void spmm_fused(const long long* __restrict__ esrc,
                           const long long* __restrict__ edst,
                           const float* __restrict__ ew,
                           const float* __restrict__ xW,
                           const float* __restrict__ emb,
                           float* __restrict__ acc1,
                           float* __restrict__ acc2, int E) {
    int t = blockIdx.x * 256 + threadIdx.x;
    int e = t >> 5;
    if (e >= E) return;
    int lane = t & 31;

    long long s = esrc[e];
    long long d = edst[e];
    float wt = ew[e];

    size_t so  = (size_t)s * DD + lane * 4;
    size_t dof = (size_t)d * DD + lane * 4;

    float4 xv = *(const float4*)(xW + so);
    float4 ev = *(const float4*)(emb + so);

    float* a1 = acc1 + dof;
    float* a2 = acc2 + dof;
    atomAddF32(a1 + 0, wt * xv.x);
    atomAddF32(a1 + 1, wt * xv.y);
    atomAddF32(a1 + 2, wt * xv.z);
    atomAddF32(a1 + 3, wt * xv.w);
    atomAddF32(a2 + 0, wt * ev.x);
    atomAddF32(a2 + 1, wt * ev.y);
    atomAddF32(a2 + 2, wt * ev.z);
    atomAddF32(a2 + 3, wt * ev.w);
}

// ---------------------------------------------------------------------------
// 6) out = acc2 @ Wk + out   (WMMA f32, C seeded from out = gcn branch)
//    Wk is row-major [k][j] -> B fragment elements come from rows
//    k+2*half and k+2*half+1 at column (nt*16+lid).
// ---------------------------------------------------------------------------
__global__ void gemm_out(const float* __restrict__ A2,
                         const float* __restrict__ Wk,
                         float* __restrict__ Out, int ntiles) {
    int wave = blockIdx.x * (blockDim.x >> 5) + (threadIdx.x >> 5);
    if (wave >= ntiles) return;
    int lane = threadIdx.x & 31;
    int mt = wave >> 3;
    int nt = wave & 7;
    int half = lane >> 4, lid = lane & 15;
    int col = nt * 16 + lid;

    float* Op = Out + (size_t)(mt * 16 + 8 * half) * DD + col;
    v8f c;
#pragma unroll
    for (int v = 0; v < 8; ++v)
        c[v] = Op[(size_t)v * DD];

    const float* Ap = A2 + (size_t)(mt * 16 + lid) * DD + 2 * half;
#pragma unroll
    for (int k = 0; k < DD; k += 4) {
        v2f a = *(const v2f*)(Ap + k);
        v2f b;
        b.x = Wk[(size_t)(k + 2 * half + 0) * DD + col];
        b.y = Wk[(size_t)(k + 2 * half + 1) * DD + col];
        c = __builtin_amdgcn_wmma_f32_16x16x4_f32(false, a, false, b,
                                                  (short)0, c, false, false);
    }
#pragma unroll
    for (int v = 0; v < 8; ++v)
        Op[(size_t)v * DD] = c[v];
}

// ---------------------------------------------------------------------------
extern "C" void kernel_launch(void* const* d_in, const int* in_sizes, int n_in,
                              void* d_out, int out_size, void* d_ws, size_t ws_size,
                              hipStream_t stream) {
    const float*     features = (const float*)d_in[0];
    const long long* eidx     = (const long long*)d_in[1];   // int64 [2, E]
    const float*     ew       = (const float*)d_in[2];
    const float*     gw       = (const float*)d_in[3];
    const float*     bias     = (const float*)d_in[4];
    const float*     Fp       = (const float*)d_in[5];
    const float*     emb      = (const float*)d_in[6];
    float*           out      = (float*)d_out;

    const int N = in_sizes[0] / DD;     // 100000
    const int E = in_sizes[2];          // 3,200,000
    const long long* esrc = eidx;
    const long long* edst = eidx + E;

    // workspace layout (floats): xW[N*D] | acc2[N*D] | Wk[D*D] | ss[1]
    float* xW   = (float*)d_ws;
    float* acc2 = xW + (size_t)N * DD;
    float* Wk   = acc2 + (size_t)N * DD;
    float* ss   = Wk + DD * DD;

    const int total = N * DD;
    init_acc<<<(total + 255) / 256, 256, 0, stream>>>(bias, out, acc2, ss, total);
    wmat  <<<(DD * DD) / 256, 256, 0, stream>>>(Fp, Wk, ss);
    wscale<<<(DD * DD) / 256, 256, 0, stream>>>(Wk, ss);

    const int tiles = (N / 16) * (DD / 16);        // 50000 (N divides by 16)
    gemm_xw<<<(tiles + 3) / 4, 128, 0, stream>>>(features, gw, xW, tiles);

    long long sthreads = (long long)E * 32;        // one wave per edge
    spmm_fused<<<(unsigned)((sthreads + 255) / 256), 256, 0, stream>>>(
        esrc, edst, ew, xW, emb, out, acc2, E);

    gemm_out<<<(tiles + 3) / 4, 128, 0, stream>>>(acc2, Wk, out, tiles);
}